// AttentionFlow_3332894622229
// MI455X (gfx1250) — compile-verified
//
#include <hip/hip_runtime.h>
#include <hip/hip_bf16.h>
#include <stdint.h>

#define AS1 __attribute__((address_space(1)))
#define AS3 __attribute__((address_space(3)))

#if defined(__has_builtin)
#  if __has_builtin(__builtin_amdgcn_global_load_async_to_lds_b128)
#    define HAVE_ASYNC_LDS 1
#  endif
#  if __has_builtin(__builtin_amdgcn_s_wait_asynccnt)
#    define HAVE_WAIT_ASYNC 1
#  endif
#  if __has_builtin(__builtin_amdgcn_wmma_f32_16x16x4_f32)
#    define HAVE_WMMA_F32X4 1
#  endif
#endif

typedef __attribute__((ext_vector_type(2))) float v2f;
typedef __attribute__((ext_vector_type(4))) float v4f;
typedef __attribute__((ext_vector_type(8))) float v8f;
typedef __attribute__((ext_vector_type(4))) int   v4i;

// ---------------------------------------------------------------------------
// Kernel 1: per-edge L1 losses. One wave32 per edge, lane l owns 4 contiguous
// floats of the D=128 row (b128 loads). Streams rel/hidden with NT hints
// (512 MB once-through); query tables (512 KB) stay L2-resident.
// ---------------------------------------------------------------------------
__global__ void __launch_bounds__(256)
losses_kernel(const int* __restrict__ edges,
              const float* __restrict__ hidden,   // visited_node_representation[2]
              const float* __restrict__ rel,
              const float* __restrict__ qsrc,
              const float* __restrict__ qrel,
              float* __restrict__ rel_loss,
              float* __restrict__ tri_loss,
              int E) {
  const int lane = threadIdx.x & 31;
  const int wave = threadIdx.x >> 5;
  const int e = blockIdx.x * 8 + wave;
  if (e >= E) return;

  const int q = edges[(size_t)e * 9 + 8];

  const v4f r  = __builtin_nontemporal_load((const v4f*)(rel    + (size_t)e * 128) + lane);
  const v4f h  = __builtin_nontemporal_load((const v4f*)(hidden + (size_t)e * 128) + lane);
  const v4f qs = ((const v4f*)(qsrc + (size_t)q * 128))[lane];
  const v4f qr = ((const v4f*)(qrel + (size_t)q * 128))[lane];

  float t  = fabsf(qs.x + r.x - h.x) + fabsf(qs.y + r.y - h.y)
           + fabsf(qs.z + r.z - h.z) + fabsf(qs.w + r.w - h.w);
  float rl = fabsf(qr.x - r.x) + fabsf(qr.y - r.y)
           + fabsf(qr.z - r.z) + fabsf(qr.w - r.w);

#pragma unroll
  for (int off = 16; off > 0; off >>= 1) {
    t  += __shfl_xor(t,  off, 32);
    rl += __shfl_xor(rl, off, 32);
  }
  if (lane == 0) {
    rel_loss[e] = rl;   // -> d_out[0..E), overwritten by score later (same tile/wave)
    tri_loss[e] = t;    // -> d_out[E..2E) staging, consumed before reg copy
  }
}

// ---------------------------------------------------------------------------
// Kernel 2: target_score = [rel, sample, triple] @ (-|w|), done as a real
// matmul on the matrix pipe: one V_WMMA_F32_16X16X4_F32 per 16 edges.
// A (16x4, ISA layout): lanes 0-15 hold K=0(rel),K=1(sample);
//                       lanes 16-31 hold K=2(triple),K=3(0).
// B (4x16) broadcasts -|w_k| across all N columns, so every column of D is
// the 16-score vector; lane 0 holds rows 0-7, lane 16 rows 8-15 (N=0).
// In-place: reads rel_loss from `score`, writes score back (same wave/tile).
// ---------------------------------------------------------------------------
__global__ void __launch_bounds__(256)
combine_wmma_kernel(const float* __restrict__ stepw,
                    const float* __restrict__ sample,
                    float* __restrict__ score,       // in: rel_loss, out: target_score
                    const float* __restrict__ tri,
                    int ntiles, int E) {
  const int lane = threadIdx.x & 31;
  const int wave = threadIdx.x >> 5;
  const int t = blockIdx.x * 8 + wave;
  if (t >= ntiles) return;
  const int base = t * 16;
  const int m = lane & 15;
  int e = base + m;
  if (e >= E) e = E - 1;  // clamp reads; stores are guarded

  const float w0 = -fabsf(stepw[0]);
  const float w1 = -fabsf(stepw[1]);
  const float w2 = -fabsf(stepw[2]);

#ifdef HAVE_WMMA_F32X4
  const bool hi = lane >= 16;
  v2f a, b;
  a.x = hi ? tri[e]  : score[e];
  a.y = hi ? 0.0f    : sample[e];
  b.x = hi ? w2 : w0;
  b.y = hi ? 0.0f : w1;  // K=3 row is zero
  v8f c = {};
  c = __builtin_amdgcn_wmma_f32_16x16x4_f32(false, a, false, b, (short)0, c, false, false);
  if (lane == 0) {
#pragma unroll
    for (int i = 0; i < 8; ++i)
      if (base + i < E) score[base + i] = c[i];
  } else if (lane == 16) {
#pragma unroll
    for (int i = 0; i < 8; ++i)
      if (base + 8 + i < E) score[base + 8 + i] = c[i];
  }
#else
  if (lane < 16 && base + m < E)
    score[e] = w0 * score[e] + w1 * sample[e] + w2 * tri[e];
#endif
}

// ---------------------------------------------------------------------------
// Kernel 3: visited_nodes_reg pass-through (b128 NT copy).
// ---------------------------------------------------------------------------
__global__ void __launch_bounds__(256)
copy_reg_kernel(const float* __restrict__ src, float* __restrict__ dst, int n) {
  const int gi = blockIdx.x * blockDim.x + threadIdx.x;
  const int n4 = n >> 2;
  for (int j = gi; j < n4; j += gridDim.x * blockDim.x) {
    v4f v = __builtin_nontemporal_load((const v4f*)src + j);
    __builtin_nontemporal_store(v, (v4f*)dst + j);
  }
  const int rem = n & 3;
  if (gi < rem) dst[(n4 << 2) + gi] = src[(n4 << 2) + gi];
}

// ---------------------------------------------------------------------------
// Kernel 4: per-group top-K (K=10 of EPG=1000) + edge gather.
// Stages the group's scores into LDS via gfx1250 async-load-to-LDS
// (ASYNCcnt path) when available; then K sequential argmax passes with
// lowest-index tie-break (matches jax.lax.top_k descending order).
// ---------------------------------------------------------------------------
__global__ void __launch_bounds__(256)
topk_kernel(const float* __restrict__ score,
            const int* __restrict__ edges,
            int* __restrict__ pruned,
            int* __restrict__ oidx,
            int EPG, int K) {
  __shared__ __align__(16) float s[1024];
  __shared__ float rv[256];
  __shared__ int   ri[256];
  __shared__ int   topidx[32];

  const int b = blockIdx.x, tid = threadIdx.x;
  const float* gs = score + (size_t)b * EPG;

#ifdef HAVE_ASYNC_LDS
  if ((EPG & 3) == 0 && EPG <= 1024) {
    const int nvec = EPG >> 2;
    if (tid < nvec) {
      __builtin_amdgcn_global_load_async_to_lds_b128(
          (AS1 v4i*)(gs + tid * 4), (AS3 v4i*)(&s[tid * 4]), 0, 0);
    }
#ifdef HAVE_WAIT_ASYNC
    __builtin_amdgcn_s_wait_asynccnt(0);
#else
    asm volatile("s_wait_asynccnt 0x0" ::: "memory");
#endif
  } else
#endif
  {
    for (int i = tid; i < EPG; i += 256) s[i] = gs[i];
  }
  __syncthreads();

  for (int k = 0; k < K; ++k) {
    float bv = -__builtin_inff();
    int   bi = 0x7fffffff;
    for (int i = tid; i < EPG; i += 256) {
      float v = s[i];
      if (v > bv) { bv = v; bi = i; }   // ascending i => lowest index kept on ties
    }
    rv[tid] = bv; ri[tid] = bi;
    __syncthreads();
    for (int st = 128; st > 0; st >>= 1) {
      if (tid < st) {
        float v = rv[tid + st]; int ix = ri[tid + st];
        if (v > rv[tid] || (v == rv[tid] && ix < ri[tid])) { rv[tid] = v; ri[tid] = ix; }
      }
      __syncthreads();
    }
    if (tid == 0) { topidx[k] = ri[0]; s[ri[0]] = -__builtin_inff(); }
    __syncthreads();
  }

  if (tid < K) oidx[(size_t)b * K + tid] = b * EPG + topidx[tid];
  if (tid < K * 9) {
    const int r = tid / 9, c = tid % 9;
    const int orig = b * EPG + topidx[r];
    pruned[((size_t)b * K + r) * 9 + c] = edges[(size_t)orig * 9 + c];
  }
}

// ---------------------------------------------------------------------------
// d_out layout (1,050,000 elems): [0,E) target_score f32 | [E,2E) reg f32 |
// [2E, 2E+B*K*9) pruned_edges i32 | then orig_indices i32. The [E,2E) slot
// doubles as triple_loss staging before the reg copy lands there.
// ---------------------------------------------------------------------------
extern "C" void kernel_launch(void* const* d_in, const int* in_sizes, int n_in,
                              void* d_out, int out_size, void* d_ws, size_t ws_size,
                              hipStream_t stream) {
  const float* stepw  = (const float*)d_in[0];
  const float* vreg   = (const float*)d_in[2];
  const float* sample = (const float*)d_in[4];
  const int*   edges  = (const int*)d_in[5];
  const float* vrep   = (const float*)d_in[6];
  const float* rel    = (const float*)d_in[7];
  const float* qsrc   = (const float*)d_in[8];
  const float* qrel   = (const float*)d_in[9];

  const int E   = in_sizes[1];          // 500000
  const int D   = in_sizes[7] / E;      // 128 (losses kernel assumes 128)
  const int B   = in_sizes[8] / D;      // 500
  const int EPG = E / B;                // 1000
  const int K   = 10;                   // max_edges (python scalar in reference)

  const float* hidden = vrep + (size_t)2 * E * D;

  float* score  = (float*)d_out;
  float* tri    = score + E;
  int*   pruned = (int*)(score + (size_t)2 * E);
  int*   oidx   = pruned + (size_t)B * K * 9;

  losses_kernel<<<(E + 7) / 8, 256, 0, stream>>>(edges, hidden, rel, qsrc, qrel,
                                                 score, tri, E);

  const int ntiles = (E + 15) / 16;
  combine_wmma_kernel<<<(ntiles + 7) / 8, 256, 0, stream>>>(stepw, sample, score,
                                                            tri, ntiles, E);

  copy_reg_kernel<<<1024, 256, 0, stream>>>(vreg, tri, E);

  topk_kernel<<<B, 256, 0, stream>>>(score, edges, pruned, oidx, EPG, K);
}